// EpisodicMemory_66383014527218
// MI455X (gfx1250) — compile-verified
//
#include <hip/hip_runtime.h>
#include <math.h>

#define BATCH      4096
#define NUM_SLOTS  16384
#define KEY_DIM    64
#define VALUE_DIM  128
#define TOP_K      32
#define TEMP       8.0f
#define EPS_V      1e-6f
#define NEG_INF_V  (-1e9f)
#define USAGE_EMA_V 0.05f

#define BM 64        // query rows per block (sim kernel)
#define BN 128       // slot cols per block (sim kernel)
#define LDA 68       // padded LDS row stride (floats): 68 % 64 == 4 -> conflict-free b64 reads
#define UCHUNK 16    // batch chunks for usage partial sums

typedef float v2f __attribute__((ext_vector_type(2)));
typedef float v8f __attribute__((ext_vector_type(8)));

// ---------------------------------------------------------------------------
// Kernel A: row-wise L2 normalize (dim = 64). One wave32 per row.
// ---------------------------------------------------------------------------
__global__ void em_normalize_rows(const float* __restrict__ in,
                                  float* __restrict__ out, int nrows) {
    int wave = (blockIdx.x * blockDim.x + threadIdx.x) >> 5;
    int lane = threadIdx.x & 31;
    if (wave >= nrows) return;
    const float* r = in + (size_t)wave * KEY_DIM;
    float a = r[lane];
    float b = r[lane + 32];
    float s = a * a + b * b;
    #pragma unroll
    for (int m = 16; m >= 1; m >>= 1) s += __shfl_xor(s, m, 32);
    float scale = 1.0f / fmaxf(sqrtf(s), EPS_V);
    float* o = out + (size_t)wave * KEY_DIM;
    o[lane]      = a * scale;
    o[lane + 32] = b * scale;
}

// ---------------------------------------------------------------------------
// Kernel B: sim = qn @ kn^T via V_WMMA_F32_16X16X4_F32, masked by alive,
// written dense into the weights output region.
//
// Block = 256 threads (8 waves), output tile 64x128.
//  - A (64x64 of qn) staged once per block in LDS (stride-68 padding).
//  - Each wave owns one 16-col slot group: its B operands (16 v2f = 32 VGPRs)
//    are loaded once from global and reused across 4 M-subtiles.
//  - Per wave: 4 subtiles x 16 chained K=4 WMMAs = 64 v_wmma ops.
//
// WMMA f32 16x16x4 layout:
//  A: lanes 0-15 hold M=lane with K-pair {k,k+1}; lanes 16-31 K-pair {k+2,k+3}
//  B: lanes 0-15 hold N=lane, same K-pair split; B[k][n] = kn[n][k]
//  C/D: lane holds col N=lane%16; VGPR r holds row M = r + 8*(lane/16)
// ---------------------------------------------------------------------------
__global__ __launch_bounds__(256) void em_sim_wmma(
        const float* __restrict__ qn, const float* __restrict__ kn,
        const int* __restrict__ alive, float* __restrict__ sim_out) {
    __shared__ float sA[BM][LDA];

    const int tid    = threadIdx.x;
    const int waveId = tid >> 5;
    const int lane   = tid & 31;
    const int m0 = blockIdx.x * BM;
    const int n0 = blockIdx.y * BN + waveId * 16;
    const int half = lane >> 4;   // selects K-pair
    const int l    = lane & 15;   // row (A) / col (B) index

    // ---- cooperative stage of A tile: thread -> row tid/4, 16-col chunk ----
    {
        const int r = tid >> 2;
        const int c = (tid & 3) << 4;
        const float* src = qn + (size_t)(m0 + r) * KEY_DIM + c;
        float4 x0 = *(const float4*)(src + 0);
        float4 x1 = *(const float4*)(src + 4);
        float4 x2 = *(const float4*)(src + 8);
        float4 x3 = *(const float4*)(src + 12);
        *(float4*)&sA[r][c + 0]  = x0;
        *(float4*)&sA[r][c + 4]  = x1;
        *(float4*)&sA[r][c + 8]  = x2;
        *(float4*)&sA[r][c + 12] = x3;
    }

    // ---- B operands into registers (reused across 4 M-subtiles) ----
    v2f breg[16];
    {
        const float* brow = kn + (size_t)(n0 + l) * KEY_DIM + 2 * half;
        #pragma unroll
        for (int kk = 0; kk < 16; ++kk) breg[kk] = *(const v2f*)(brow + 4 * kk);
    }
    const bool al = alive[n0 + l] != 0;

    __syncthreads();

    #pragma unroll
    for (int mt = 0; mt < 4; ++mt) {
        v8f c8 = {0.f, 0.f, 0.f, 0.f, 0.f, 0.f, 0.f, 0.f};
        const float* arow = &sA[mt * 16 + l][2 * half];
        #pragma unroll
        for (int kk = 0; kk < 16; ++kk) {
            v2f a = *(const v2f*)(arow + 4 * kk);   // ds_load_b64, conflict-free
            c8 = __builtin_amdgcn_wmma_f32_16x16x4_f32(
                    false, a, false, breg[kk], (short)0, c8, false, false);
        }
        #pragma unroll
        for (int r = 0; r < 8; ++r) {
            float v = al ? c8[r] : NEG_INF_V;
            sim_out[(size_t)(m0 + mt * 16 + r + 8 * half) * NUM_SLOTS + (n0 + l)] = v;
        }
    }
}

// ---------------------------------------------------------------------------
// Kernel C: per query row — top-32 (value desc, index asc tie-break),
// softmax(*8), zero + scatter the row, and compute the weighted value read.
// One 256-thread block per row; sim row lives in (and is rewritten into)
// the weights output region.
// ---------------------------------------------------------------------------
__global__ void em_topk_softmax(float* __restrict__ weights,
                                const float* __restrict__ mem_values,
                                float* __restrict__ values_out) {
    __shared__ float redv[8];
    __shared__ int   redi[8];
    __shared__ float topv[TOP_K];
    __shared__ int   topi[TOP_K];
    __shared__ float wts[TOP_K];

    const int b    = blockIdx.x;
    const int tid  = threadIdx.x;        // 0..255
    const int lane = tid & 31;
    const int wid  = tid >> 5;
    float* wrow = weights + (size_t)b * NUM_SLOTS;

    // Per-thread cached local max over strided indices tid, tid+256, ...
    // Ascending scan with strict '>' keeps the smallest index on ties.
    float lv = -INFINITY; int li = tid;
    for (int i = tid; i < NUM_SLOTS; i += 256) {
        float v = wrow[i];
        if (v > lv) { lv = v; li = i; }
    }

    for (int t = 0; t < TOP_K; ++t) {
        float v = lv; int i = li;
        #pragma unroll
        for (int m = 16; m >= 1; m >>= 1) {
            float ov = __shfl_xor(v, m, 32);
            int   oi = __shfl_xor(i, m, 32);
            if (ov > v || (ov == v && oi < i)) { v = ov; i = oi; }
        }
        if (lane == 0) { redv[wid] = v; redi[wid] = i; }
        __syncthreads();
        if (tid == 0) {
            float bv = redv[0]; int bi = redi[0];
            #pragma unroll
            for (int w = 1; w < 8; ++w) {
                if (redv[w] > bv || (redv[w] == bv && redi[w] < bi)) {
                    bv = redv[w]; bi = redi[w];
                }
            }
            topv[t] = bv; topi[t] = bi;
        }
        __syncthreads();
        const int gi = topi[t];
        if ((gi & 255) == tid) {   // owner knocks it out and rescans its stripe
            wrow[gi] = -INFINITY;
            lv = -INFINITY; li = tid;
            for (int i2 = tid; i2 < NUM_SLOTS; i2 += 256) {
                float v2 = wrow[i2];
                if (v2 > lv) { lv = v2; li = i2; }
            }
        }
        __syncthreads();
    }

    // Softmax over the 32 top values (topv[0] is the max).
    if (tid < 32) {
        float e = expf((topv[tid] - topv[0]) * TEMP);
        float s = e;
        #pragma unroll
        for (int m = 16; m >= 1; m >>= 1) s += __shfl_xor(s, m, 32);
        wts[tid] = e / s;
    }
    __syncthreads();

    // Zero the dense row, then scatter the 32 sparse weights.
    for (int i = tid; i < NUM_SLOTS; i += 256) wrow[i] = 0.0f;
    __syncthreads();
    if (tid < TOP_K) wrow[topi[tid]] = wts[tid];

    // values[b] = sum_j wts[j] * mem_values[topi[j]]
    if (tid < VALUE_DIM) {
        float acc = 0.0f;
        #pragma unroll
        for (int j = 0; j < TOP_K; ++j) {
            acc += wts[j] * mem_values[(size_t)topi[j] * VALUE_DIM + tid];
        }
        values_out[(size_t)b * VALUE_DIM + tid] = acc;
    }
}

// ---------------------------------------------------------------------------
// Kernel D (parallel path): partial column sums over batch chunks.
// Deterministic: fixed chunking, fixed serial order within each chunk.
// ---------------------------------------------------------------------------
__global__ void em_usage_partial(const float* __restrict__ weights,
                                 float* __restrict__ partials) {
    int s = blockIdx.x * blockDim.x + threadIdx.x;
    int c = blockIdx.y;
    const int span = BATCH / UCHUNK;       // 256
    const int b0 = c * span;
    float acc = 0.0f;
    for (int b = 0; b < span; ++b)
        acc += weights[(size_t)(b0 + b) * NUM_SLOTS + s];
    partials[(size_t)c * NUM_SLOTS + s] = acc;
}

__global__ void em_usage_final(const float* __restrict__ partials,
                               const float* __restrict__ usage,
                               const float* __restrict__ access_count,
                               float* __restrict__ new_usage,
                               float* __restrict__ new_access) {
    int s = blockIdx.x * blockDim.x + threadIdx.x;
    float acc = 0.0f;
    #pragma unroll
    for (int c = 0; c < UCHUNK; ++c)
        acc += partials[(size_t)c * NUM_SLOTS + s];
    float delta = acc * (1.0f / (float)BATCH);
    new_usage[s]  = usage[s] + delta * USAGE_EMA_V;
    new_access[s] = access_count[s] + delta;
}

// Fallback: single-pass column reduction (used if d_ws is too small).
__global__ void em_usage_update(const float* __restrict__ weights,
                                const float* __restrict__ usage,
                                const float* __restrict__ access_count,
                                float* __restrict__ new_usage,
                                float* __restrict__ new_access) {
    int s = blockIdx.x * blockDim.x + threadIdx.x;
    if (s >= NUM_SLOTS) return;
    float acc = 0.0f;
    for (int b = 0; b < BATCH; ++b)
        acc += weights[(size_t)b * NUM_SLOTS + s];
    float delta = acc * (1.0f / (float)BATCH);
    new_usage[s]  = usage[s] + delta * USAGE_EMA_V;
    new_access[s] = access_count[s] + delta;
}

// ---------------------------------------------------------------------------
extern "C" void kernel_launch(void* const* d_in, const int* in_sizes, int n_in,
                              void* d_out, int out_size, void* d_ws, size_t ws_size,
                              hipStream_t stream) {
    const float* query        = (const float*)d_in[0];
    const float* mem_keys     = (const float*)d_in[1];
    const float* mem_values   = (const float*)d_in[2];
    const int*   alive        = (const int*)d_in[3];
    const float* usage        = (const float*)d_in[4];
    const float* access_count = (const float*)d_in[5];
    // d_in[6] = top_k scalar (compile-time TOP_K)

    float* out_values  = (float*)d_out;                          // [B, Dv]
    float* out_weights = out_values + (size_t)BATCH * VALUE_DIM;  // [B, S]
    float* out_usage   = out_weights + (size_t)BATCH * NUM_SLOTS;
    float* out_access  = out_usage + NUM_SLOTS;

    float* kn       = (float*)d_ws;                              // [S, Dk]
    float* qn       = kn + (size_t)NUM_SLOTS * KEY_DIM;          // [B, Dk]
    float* partials = qn + (size_t)BATCH * KEY_DIM;              // [UCHUNK, S]
    const size_t ws_need_partials =
        ((size_t)NUM_SLOTS * KEY_DIM + (size_t)BATCH * KEY_DIM +
         (size_t)UCHUNK * NUM_SLOTS) * sizeof(float);

    // A: normalize keys (one wave per row) and queries
    em_normalize_rows<<<(NUM_SLOTS * 32 + 255) / 256, 256, 0, stream>>>(
        mem_keys, kn, NUM_SLOTS);
    em_normalize_rows<<<(BATCH * 32 + 255) / 256, 256, 0, stream>>>(
        query, qn, BATCH);

    // B: WMMA cosine-sim (64x128 block tiles), masked, dense into weights region
    dim3 gridB(BATCH / BM, NUM_SLOTS / BN);
    em_sim_wmma<<<gridB, 256, 0, stream>>>(qn, kn, alive, out_weights);

    // C: top-k + softmax + scatter + weighted value read (one block per row)
    em_topk_softmax<<<BATCH, 256, 0, stream>>>(out_weights, mem_values, out_values);

    // D: usage / access-count update from final sparse weights
    if (ws_size >= ws_need_partials) {
        dim3 gridP(NUM_SLOTS / 256, UCHUNK);
        em_usage_partial<<<gridP, 256, 0, stream>>>(out_weights, partials);
        em_usage_final<<<NUM_SLOTS / 256, 256, 0, stream>>>(
            partials, usage, access_count, out_usage, out_access);
    } else {
        em_usage_update<<<NUM_SLOTS / 256, 256, 0, stream>>>(
            out_weights, usage, access_count, out_usage, out_access);
    }
}